// RelationalSelfAttention_26688926777864
// MI455X (gfx1250) — compile-verified
//
#include <hip/hip_runtime.h>

typedef float v2f __attribute__((ext_vector_type(2)));
typedef float v8f __attribute__((ext_vector_type(8)));

#define B_DIM 512
#define M_DIM 3
#define H_DIM 768

// ---------------------------------------------------------------------------
// Fused QKV projection. One launch computes:
//   z==0 : Q[512 x768]  = target @ Wq^T + bq
//   z==1 : K[1536x768]  = cont   @ Wk^T + bk
//   z==2 : V[1536x768]  = cont   @ Wv^T + bv
// Uses V_WMMA_F32_16X16X4_F32 (full fp32 precision).
// Block = 256 threads = 8 waves. Each wave computes a 32x64 output strip:
// 2 M-tiles x 4 N-tiles = 8 accumulators, so each B fragment feeds 2 WMMAs
// and each A fragment feeds 4 (~10.7 FLOP per operand byte from cache).
// ---------------------------------------------------------------------------
__global__ __launch_bounds__(256) void rsa_qkv_gemm(
    const float* __restrict__ target, const float* __restrict__ cont,
    const float* __restrict__ Wq, const float* __restrict__ bq,
    const float* __restrict__ Wk, const float* __restrict__ bk,
    const float* __restrict__ Wv, const float* __restrict__ bv,
    float* __restrict__ Qo, float* __restrict__ Ko, float* __restrict__ Vo)
{
  const int K = H_DIM, N = H_DIM;
  const int z = blockIdx.z;                       // which projection (uniform)
  const float* A    = (z == 0) ? target : cont;
  const float* W    = (z == 0) ? Wq : (z == 1) ? Wk : Wv;
  const float* bias = (z == 0) ? bq : (z == 1) ? bk : bv;
  float*       C    = (z == 0) ? Qo : (z == 1) ? Ko : Vo;
  const int Mrows   = (z == 0) ? B_DIM : (B_DIM * M_DIM);

  const int lane = threadIdx.x & 31;
  const int wave = threadIdx.x >> 5;              // 0..7
  const int half = lane >> 4;                     // 0 or 1 (lane half)
  const int l15  = lane & 15;

  const int rowBase = (blockIdx.x * 8 + wave) * 32;   // wave-uniform
  const int colBase = blockIdx.y * 64;
  if (rowBase >= Mrows) return;                   // uniform -> EXEC stays all-1s

  v8f acc[2][4] = {};                             // 2 M-tiles x 4 N-tiles

  // A fragment (16x4 f32): lane l15 = row; VGPR0/1 hold K = 2*half + {0,1}
  const float* aPtr = A + (size_t)(rowBase + l15) * K + 2 * half;
  // B fragment (4x16 f32): lane l15 = col n; VGPR0/1 hold rows k = 2*half + {0,1}
  // B(k,n) = W[n*K + k]  (because y = x @ W^T)
  const float* wPtr = W + (size_t)(colBase + l15) * K + 2 * half;

  for (int k0 = 0; k0 < K; k0 += 4) {
    v2f a0 = *(const v2f*)(aPtr + k0);
    v2f a1 = *(const v2f*)(aPtr + (size_t)16 * K + k0);
    v2f b[4];
    #pragma unroll
    for (int t = 0; t < 4; ++t)
      b[t] = *(const v2f*)(wPtr + (size_t)(16 * t) * K + k0);
    #pragma unroll
    for (int t = 0; t < 4; ++t) {
      acc[0][t] = __builtin_amdgcn_wmma_f32_16x16x4_f32(false, a0, false, b[t], (short)0, acc[0][t], false, false);
      acc[1][t] = __builtin_amdgcn_wmma_f32_16x16x4_f32(false, a1, false, b[t], (short)0, acc[1][t], false, false);
    }
  }

  // C/D layout: element i -> row tileRow + i + 8*half, col colBase + 16*t + l15
  #pragma unroll
  for (int mt = 0; mt < 2; ++mt) {
    #pragma unroll
    for (int t = 0; t < 4; ++t) {
      const int n = colBase + 16 * t + l15;
      const float bv2 = bias[n];
      #pragma unroll
      for (int i = 0; i < 8; ++i) {
        C[(size_t)(rowBase + 16 * mt + i + 8 * half) * N + n] = acc[mt][t][i] + bv2;
      }
    }
  }
}

// ---------------------------------------------------------------------------
// Epilogue: one wave (32 lanes) per batch element.
// kernel_v[m] = p[m,:].q ; kernel_r[m] = sum_{m2,h} (q[h]*k[m2,h]) * hmat[(m2*H+h)*3+m]
// S[i][j] = v[i,:].v[j,:] (3x3 Gram) ; context = S@g + v ; out = kernel . context
// (The [B,H,H] intermediate _xg of the reference collapses to the 3x3 Gram.)
// ---------------------------------------------------------------------------
__device__ __forceinline__ float wave_sum32(float x) {
  #pragma unroll
  for (int m = 16; m >= 1; m >>= 1) x += __shfl_xor(x, m, 32);
  return x;
}

__global__ __launch_bounds__(256) void rsa_epilogue(
    const float* __restrict__ Q, const float* __restrict__ Kc,
    const float* __restrict__ V, const float* __restrict__ p,
    const float* __restrict__ hm, const float* __restrict__ g,
    float* __restrict__ out)
{
  const int tid  = blockIdx.x * blockDim.x + threadIdx.x;
  const int b    = tid >> 5;
  const int lane = tid & 31;
  if (b >= B_DIM) return;

  const float* q  = Q  + (size_t)b * H_DIM;
  const float* kb = Kc + (size_t)b * (M_DIM * H_DIM);
  const float* vb = V  + (size_t)b * (M_DIM * H_DIM);

  float kv0=0.f, kv1=0.f, kv2=0.f;
  float kr0=0.f, kr1=0.f, kr2=0.f;
  float s00=0.f, s01=0.f, s02=0.f, s11=0.f, s12=0.f, s22=0.f;

  for (int hh = lane; hh < H_DIM; hh += 32) {
    const float qh = q[hh];
    const float k0 = kb[hh], k1 = kb[H_DIM + hh], k2 = kb[2 * H_DIM + hh];
    const float v0 = vb[hh], v1 = vb[H_DIM + hh], v2 = vb[2 * H_DIM + hh];

    kv0 += p[hh] * qh;
    kv1 += p[H_DIM + hh] * qh;
    kv2 += p[2 * H_DIM + hh] * qh;

    {
      const float xq = qh * k0;
      const float* hr = hm + (size_t)(0 * H_DIM + hh) * 3;
      kr0 += xq * hr[0]; kr1 += xq * hr[1]; kr2 += xq * hr[2];
    }
    {
      const float xq = qh * k1;
      const float* hr = hm + (size_t)(1 * H_DIM + hh) * 3;
      kr0 += xq * hr[0]; kr1 += xq * hr[1]; kr2 += xq * hr[2];
    }
    {
      const float xq = qh * k2;
      const float* hr = hm + (size_t)(2 * H_DIM + hh) * 3;
      kr0 += xq * hr[0]; kr1 += xq * hr[1]; kr2 += xq * hr[2];
    }

    s00 += v0 * v0; s01 += v0 * v1; s02 += v0 * v2;
    s11 += v1 * v1; s12 += v1 * v2; s22 += v2 * v2;
  }

  kv0 = wave_sum32(kv0); kv1 = wave_sum32(kv1); kv2 = wave_sum32(kv2);
  kr0 = wave_sum32(kr0); kr1 = wave_sum32(kr1); kr2 = wave_sum32(kr2);
  s00 = wave_sum32(s00); s01 = wave_sum32(s01); s02 = wave_sum32(s02);
  s11 = wave_sum32(s11); s12 = wave_sum32(s12); s22 = wave_sum32(s22);

  const float ker0 = kv0 + kr0, ker1 = kv1 + kr1, ker2 = kv2 + kr2;

  for (int hh = lane; hh < H_DIM; hh += 32) {
    const float v0 = vb[hh], v1 = vb[H_DIM + hh], v2 = vb[2 * H_DIM + hh];
    const float g0 = g[hh],  g1 = g[H_DIM + hh],  g2 = g[2 * H_DIM + hh];
    const float c0 = s00 * g0 + s01 * g1 + s02 * g2 + v0;   // context row 0
    const float c1 = s01 * g0 + s11 * g1 + s12 * g2 + v1;   // context row 1
    const float c2 = s02 * g0 + s12 * g1 + s22 * g2 + v2;   // context row 2
    out[(size_t)b * H_DIM + hh] = ker0 * c0 + ker1 * c1 + ker2 * c2;
  }
}

// ---------------------------------------------------------------------------
extern "C" void kernel_launch(void* const* d_in, const int* in_sizes, int n_in,
                              void* d_out, int out_size, void* d_ws, size_t ws_size,
                              hipStream_t stream) {
  const float* target = (const float*)d_in[0];   // [512, 768]
  const float* cont   = (const float*)d_in[1];   // [512, 3, 768] == [1536, 768]
  const float* Wq     = (const float*)d_in[2];
  const float* bq     = (const float*)d_in[3];
  const float* Wk     = (const float*)d_in[4];
  const float* bk     = (const float*)d_in[5];
  const float* Wv     = (const float*)d_in[6];
  const float* bv     = (const float*)d_in[7];
  const float* p      = (const float*)d_in[8];   // [3, 768]
  const float* hmat   = (const float*)d_in[9];   // [2304, 3]
  const float* g      = (const float*)d_in[10];  // [3, 768]
  float* out = (float*)d_out;                    // [512, 768]

  float* Qws = (float*)d_ws;                     // 512*768
  float* Kws = Qws + (size_t)B_DIM * H_DIM;      // 1536*768
  float* Vws = Kws + (size_t)B_DIM * M_DIM * H_DIM;

  dim3 blk(256);
  // One fused launch for all three projections:
  //  grid.x: row super-tiles of 256 rows (8 waves x 32 rows), sized for the
  //          largest problem (1536 rows -> 6); Q blocks with rowBase >= 512
  //          exit immediately (wave-uniform).
  //  grid.y: 768 cols / 64-col wave strips = 12
  //  grid.z: 0=Q, 1=K, 2=V
  rsa_qkv_gemm<<<dim3((B_DIM * M_DIM) / 256, H_DIM / 64, 3), blk, 0, stream>>>(
      target, cont, Wq, bq, Wk, bk, Wv, bv, Qws, Kws, Vws);

  // one wave per batch element: 512 waves = 64 blocks of 256 threads
  rsa_epilogue<<<dim3((B_DIM * 32) / 256), blk, 0, stream>>>(Qws, Kws, Vws, p, hmat, g, out);
}